// TorchGeoGNN_29257317220812
// MI455X (gfx1250) — compile-verified
//
#include <hip/hip_runtime.h>

typedef float v2f __attribute__((ext_vector_type(2)));
typedef float v8f __attribute__((ext_vector_type(8)));

#define D 128
#define BLK_ROWS 128
#define A_STRIDE 132    // floats; keeps float4 stores 16B aligned, spreads banks
#define WP_STRIDE 264   // floats per K-pair row (128 cols * 2 + 8 pad)

// ---------------------------------------------------------------------------
// Degree / normalization kernels
// ---------------------------------------------------------------------------
__global__ void gcn_deg_init(float* __restrict__ deg, int n_nodes) {
    int i = blockIdx.x * blockDim.x + threadIdx.x;
    if (i < n_nodes) deg[i] = 1.0f;  // self-loop contribution
}

__global__ void gcn_deg_count(const long long* __restrict__ dst,
                              float* __restrict__ deg, int n_edges) {
    int e = blockIdx.x * blockDim.x + threadIdx.x;
    if (e < n_edges) atomicAdd(&deg[(int)dst[e]], 1.0f);
}

__global__ void gcn_rsqrt(float* __restrict__ deg, int n_nodes) {
    int i = blockIdx.x * blockDim.x + threadIdx.x;
    if (i < n_nodes) deg[i] = rsqrtf(deg[i]);  // deg >= 1 always (self-loops)
}

// ---------------------------------------------------------------------------
// Fused GEMM + self-loop epilogue:
//   h_out[r]   = act(A[r] + bias) @ W          (act = ReLU if do_relu)
//   agg_out[r] = h_out[r] * dis[r]^2           (initializes aggregation buf)
// One block = 128 rows x 128 cols, 8 wave32, WMMA f32 16x16x4.
// W staged in LDS in K-pair-interleaved layout so every B fragment is one
// contiguous ds_load_b64 into an even-aligned VGPR pair (no repacking movs).
// ---------------------------------------------------------------------------
__global__ __launch_bounds__(256) void gcn_gemm_wmma(
    const float* __restrict__ A, const float* __restrict__ W,
    const float* __restrict__ dis,
    float* __restrict__ h_out, float* __restrict__ agg_out,
    int n_rows, const float* __restrict__ bias, int do_relu)
{
    __shared__ float As[BLK_ROWS * A_STRIDE];
    __shared__ float Wp[(D / 2) * WP_STRIDE];

    const int t    = threadIdx.x;
    const int row0 = blockIdx.x * BLK_ROWS;

    // Cooperative load of W (128x128 f32) into LDS, interleaving K-pairs:
    //   Wp[(k>>1)*WP_STRIDE + n*2 + (k&1)] = W[k][n]
    // Each thread handles a 4-column chunk of one K-pair row.
#pragma unroll
    for (int i = 0; i < 8; ++i) {
        const int pairIdx = i * 256 + t;          // 0 .. 2047
        const int pr = pairIdx >> 5;              // K-pair row: 0 .. 63
        const int cc = (pairIdx & 31) * 4;        // column chunk base
        const float4 w0 = *(const float4*)(W + (2 * pr + 0) * D + cc);
        const float4 w1 = *(const float4*)(W + (2 * pr + 1) * D + cc);
        float* p = &Wp[pr * WP_STRIDE + cc * 2];
        *(float4*)(p + 0) = make_float4(w0.x, w1.x, w0.y, w1.y);
        *(float4*)(p + 4) = make_float4(w0.z, w1.z, w0.w, w1.w);
    }
    // Cooperative load of A tile with fused (bias, ReLU) epilogue of the
    // previous layer; zero-fill rows beyond n_rows.
#pragma unroll
    for (int i = 0; i < 16; ++i) {
        const int linear = i * 1024 + t * 4;
        const int r = linear >> 7;
        const int c = linear & 127;
        float4 a4 = make_float4(0.f, 0.f, 0.f, 0.f);
        const int grow = row0 + r;
        if (grow < n_rows) a4 = *(const float4*)(A + (size_t)grow * D + c);
        if (bias) {
            a4.x += bias[c + 0]; a4.y += bias[c + 1];
            a4.z += bias[c + 2]; a4.w += bias[c + 3];
        }
        if (do_relu) {
            a4.x = fmaxf(a4.x, 0.f); a4.y = fmaxf(a4.y, 0.f);
            a4.z = fmaxf(a4.z, 0.f); a4.w = fmaxf(a4.w, 0.f);
        }
        *(float4*)(&As[r * A_STRIDE + c]) = a4;
    }
    __syncthreads();

    const int wave  = t >> 5;
    const int lane  = t & 31;
    const int half  = lane >> 4;   // 0: lanes 0-15, 1: lanes 16-31
    const int idx16 = lane & 15;   // M index for A, N index for B/C/D
    const int mrow  = wave * 16 + idx16;

    v8f acc[8] = {};

    for (int k = 0; k < D; k += 4) {
        // A fragment 16x4: lane holds A[mrow][k+2*half], A[mrow][k+2*half+1]
        const v2f a = *(const v2f*)(&As[mrow * A_STRIDE + k + 2 * half]);
        // B fragments 4x16: lane needs W[k+2*half][n], W[k+2*half+1][n],
        // contiguous in the K-pair layout -> single b64 per fragment.
        const int prow = (k >> 1) + half;
        const float* wp = &Wp[prow * WP_STRIDE + idx16 * 2];
#pragma unroll
        for (int nt = 0; nt < 8; ++nt) {
            const v2f b = *(const v2f*)(wp + nt * 32);
            acc[nt] = __builtin_amdgcn_wmma_f32_16x16x4_f32(
                /*neg_a=*/false, a, /*neg_b=*/false, b,
                /*c_mod=*/(short)0, acc[nt],
                /*reuse_a=*/false, /*reuse_b=*/false);
        }
    }

    // C/D layout: VGPR v holds row (v + 8*half), column idx16.
    // Fused epilogue: also initialize aggregation buffer with the self-loop
    // term h * dis^2, saving a full extra pass over h.
#pragma unroll
    for (int v = 0; v < 8; ++v) {
        const int grow = row0 + wave * 16 + half * 8 + v;
        if (grow < n_rows) {
            float s = dis[grow];
            s = s * s;
            const size_t base = (size_t)grow * D + idx16;
#pragma unroll
            for (int nt = 0; nt < 8; ++nt) {
                const float val = acc[nt][v];
                h_out[base + nt * 16]   = val;
                agg_out[base + nt * 16] = val * s;
            }
        }
    }
}

// ---------------------------------------------------------------------------
// Edge scatter: out[dst] += h[src] * dis[src]*dis[dst]
// One wave per edge, 4 floats per lane, fp32 atomics (L2-resident: 51 MB
// feature matrix < 192 MB L2). Grid-stride over edges; prefetch the next
// edge's source row to hide data-dependent gather latency.
// ---------------------------------------------------------------------------
__global__ void gcn_edge_scatter(const float* __restrict__ h,
                                 const float* __restrict__ dis,
                                 const long long* __restrict__ src,
                                 const long long* __restrict__ dst,
                                 float* __restrict__ out, int n_edges) {
    const int gid    = blockIdx.x * blockDim.x + threadIdx.x;
    const int lane   = threadIdx.x & 31;
    const int wid    = gid >> 5;
    const int nwaves = (gridDim.x * blockDim.x) >> 5;
    for (int e = wid; e < n_edges; e += nwaves) {
        const int en = e + nwaves;
        if (en < n_edges) {
            const int sn = (int)src[en];
            __builtin_prefetch(h + (size_t)sn * D + lane * 4, 0, 1);
        }
        const int s = (int)src[e];
        const int d = (int)dst[e];
        const float norm = dis[s] * dis[d];
        const float4 hv = *(const float4*)(h + (size_t)s * D + lane * 4);
        float* o = out + (size_t)d * D + lane * 4;
        atomicAdd(o + 0, hv.x * norm);
        atomicAdd(o + 1, hv.y * norm);
        atomicAdd(o + 2, hv.z * norm);
        atomicAdd(o + 3, hv.w * norm);
    }
}

// ---------------------------------------------------------------------------
// Final epilogue: out += b2 (no ReLU on the last layer)
// ---------------------------------------------------------------------------
__global__ void gcn_bias_add(float* __restrict__ out,
                             const float* __restrict__ bias, int n_elems) {
    const int i = blockIdx.x * blockDim.x + threadIdx.x;
    if (i < n_elems) out[i] += bias[i & (D - 1)];
}

// ---------------------------------------------------------------------------
extern "C" void kernel_launch(void* const* d_in, const int* in_sizes, int n_in,
                              void* d_out, int out_size, void* d_ws, size_t ws_size,
                              hipStream_t stream) {
    (void)n_in; (void)out_size; (void)ws_size;

    const float*     x  = (const float*)d_in[0];
    const long long* ei = (const long long*)d_in[1];  // int64 per reference
    const float*     W1 = (const float*)d_in[2];
    const float*     b1 = (const float*)d_in[3];
    const float*     W2 = (const float*)d_in[4];
    const float*     b2 = (const float*)d_in[5];
    float*           out = (float*)d_out;

    const int N = in_sizes[0] / D;
    const int E = in_sizes[1] / 2;

    const long long* src = ei;       // edge_index[0]
    const long long* dst = ei + E;   // edge_index[1]

    // Workspace: dis[N] | h[N*D] | agg[N*D]   (~103 MB total)
    float* dis = (float*)d_ws;
    float* h   = dis + N;
    float* agg = h + (size_t)N * D;

    const int T = 256;
    const int gN    = (N + T - 1) / T;
    const int gE    = (E + T - 1) / T;
    const int gGemm = (N + BLK_ROWS - 1) / BLK_ROWS;
    const int gElem = ((N * D) + T - 1) / T;
    const int gScat = 8192;                      // grid-stride over edges

    // deg -> dis = rsqrt(deg)
    gcn_deg_init <<<gN, T, 0, stream>>>(dis, N);
    gcn_deg_count<<<gE, T, 0, stream>>>(dst, dis, E);
    gcn_rsqrt    <<<gN, T, 0, stream>>>(dis, N);

    // Layer 1: h = x @ W1 ; agg = h*dis^2 (self-loop) ; += edge messages
    gcn_gemm_wmma   <<<gGemm, T, 0, stream>>>(x, W1, dis, h, agg, N, nullptr, 0);
    gcn_edge_scatter<<<gScat, T, 0, stream>>>(h, dis, src, dst, agg, E);

    // Layer 2: h = relu(agg + b1) @ W2 ; out = h*dis^2 ; += edges ; += b2
    gcn_gemm_wmma   <<<gGemm, T, 0, stream>>>(agg, W2, dis, h, out, N, b1, 1);
    gcn_edge_scatter<<<gScat, T, 0, stream>>>(h, dis, src, dst, out, E);
    gcn_bias_add    <<<gElem, T, 0, stream>>>(out, b2, N * D);
}